// SingleCurveDeformLayer_90168543412834
// MI455X (gfx1250) — compile-verified
//
#include <hip/hip_runtime.h>

typedef __attribute__((ext_vector_type(2))) float v2f;
typedef __attribute__((ext_vector_type(8))) float v8f;

#define TAU_STEP 0.5f
#define LOG2E 1.44269504088896340736f

// Build land = concat(source, target) per batch into workspace.
__global__ void pack_land_kernel(const float2* __restrict__ src,
                                 const float2* __restrict__ tgt,
                                 float2* __restrict__ land,
                                 int n_src, int n_tgt, int total) {
  int idx = blockIdx.x * blockDim.x + threadIdx.x;
  if (idx >= total) return;
  int npt = n_src + n_tgt;
  int b = idx / npt;
  int i = idx - b * npt;
  land[idx] = (i < n_src) ? src[b * n_src + i] : tgt[b * n_tgt + (i - n_src)];
}

// One Euler step of the RBF flow:
//   Out[b, iTile+m] = Xi[b, iOffset+iTile+m] + TAU * sum_j exp(-d2/sigma2) * Mom[b, j]
// Each wave32 owns 16 i-rows and accumulates with V_WMMA_F32_16X16X4_F32:
//   A(16x4) = RBF weights, B(4x16) columns {0,1} = (mom_x, mom_y), rest zero.
__launch_bounds__(256)
__global__ void rbf_step_kernel(const float2* __restrict__ Xi, int strideI, int iOffset,
                                const float2* __restrict__ Xj, int strideJ,
                                const float2* __restrict__ Mom, int strideM,
                                float2* __restrict__ Out, int strideO,
                                int nj, const float* __restrict__ sigmaV2,
                                int blocksPerBatch) {
  __shared__ float2 posT[256];        // j-point positions
  __shared__ float2 momT[256];        // j-point momenta (read back as flat floats)
  __shared__ float  shc[8 * 32 * 8];  // accumulator spill: wave x lane x vgpr

  const int lane = threadIdx.x & 31;
  const int wave = threadIdx.x >> 5;
  const int b = blockIdx.x / blocksPerBatch;
  const int tileInBatch = (blockIdx.x - b * blocksPerBatch) * 8 + wave;
  const int iTile = tileInBatch * 16;
  const int l = lane & 15;     // i-row index within tile / N column
  const int half = lane >> 4;  // K half: lanes 0-15 -> K{0,1}, 16-31 -> K{2,3}
  const int comp = l & 1;      // momentum component this lane would feed (x or y)
  const bool isMomLane = (l < 2);

  const float cexp = -LOG2E / sigmaV2[0];  // exp(-d2/s) == exp2(d2 * cexp)

  const float2 xi = Xi[(size_t)b * strideI + iOffset + iTile + l];

  v8f acc = {};  // D/C: 16x16 f32, cols 0/1 = dp_x/dp_y for rows 0..15

  const float* momF = (const float*)momT;  // momF[2*j+0]=mx, [2*j+1]=my

  for (int j0 = 0; j0 < nj; j0 += 256) {
    {  // cooperative stage of 256 j-points
      int jj = threadIdx.x;
      posT[jj] = Xj[(size_t)b * strideJ + j0 + jj];
      momT[jj] = Mom[(size_t)b * strideM + j0 + jj];
    }
    __syncthreads();

#pragma unroll 4
    for (int kk = 0; kk < 64; ++kk) {
      const int base = kk * 4 + half * 2;  // j-chunk element for this lane half
      const float2 p0 = posT[base];
      const float2 p1 = posT[base + 1];
      // Unconditional LDS reads at lane-computed addresses (no exec divergence);
      // lanes l>=2 read a valid-but-unused value and mask it to zero below.
      const float m0 = momF[2 * base + comp];
      const float m1 = momF[2 * (base + 1) + comp];

      const float dx0 = xi.x - p0.x, dy0 = xi.y - p0.y;
      const float dx1 = xi.x - p1.x, dy1 = xi.y - p1.y;
      const float d20 = dx0 * dx0 + dy0 * dy0;
      const float d21 = dx1 * dx1 + dy1 * dy1;

      v2f a, bb;
      a.x = __builtin_amdgcn_exp2f(d20 * cexp);  // A[l][2*half+0]
      a.y = __builtin_amdgcn_exp2f(d21 * cexp);  // A[l][2*half+1]
      bb.x = isMomLane ? m0 : 0.0f;  // B[2*half+0][l]
      bb.y = isMomLane ? m1 : 0.0f;  // B[2*half+1][l]

      acc = __builtin_amdgcn_wmma_f32_16x16x4_f32(
          /*neg_a=*/false, a, /*neg_b=*/false, bb,
          /*c_mod=*/(short)0, acc, /*reuse_a=*/false, /*reuse_b=*/false);
    }
    __syncthreads();
  }

  // Spill D through LDS so lane m can gather (dp_x, dp_y) of row m.
#pragma unroll
  for (int r = 0; r < 8; ++r) shc[(wave * 32 + lane) * 8 + r] = acc[r];
  __syncthreads();

  if (lane < 16) {
    const int m = lane;
    const int srcl = (m < 8) ? 0 : 16;  // lane holding N=0 for this row half
    const int r = m & 7;                // VGPR index = m % 8
    const float dpx = shc[(wave * 32 + srcl + 0) * 8 + r];
    const float dpy = shc[(wave * 32 + srcl + 1) * 8 + r];
    float2 o;
    o.x = xi.x + TAU_STEP * dpx;
    o.y = xi.y + TAU_STEP * dpy;
    Out[(size_t)b * strideO + iTile + m] = o;
  }
}

extern "C" void kernel_launch(void* const* d_in, const int* in_sizes, int n_in,
                              void* d_out, int out_size, void* d_ws, size_t ws_size,
                              hipStream_t stream) {
  const float2* momentum = (const float2*)d_in[0];  // [B, n_src, 2]
  const float2* source   = (const float2*)d_in[1];  // [B, n_src, 2]
  const float2* target   = (const float2*)d_in[2];  // [B, n_tgt, 2]
  const float*  sigmaV2  = (const float*)d_in[3];   // scalar
  float2* out = (float2*)d_out;                     // [B, n_tgt, 2]

  const int B = 4;
  const int n_src = in_sizes[1] / (B * 2);
  const int n_tgt = in_sizes[2] / (B * 2);
  const int npt = n_src + n_tgt;

  float2* land   = (float2*)d_ws;           // [B, npt] positions at t=0
  float2* shape1 = land + (size_t)B * npt;  // [B, npt] positions after step 1

  // 1) pack land = concat(source, target)
  {
    int total = B * npt;
    int blocks = (total + 255) / 256;
    pack_land_kernel<<<blocks, 256, 0, stream>>>(source, target, land, n_src, n_tgt, total);
  }

  // 2) step 1: shape1 = land + TAU * K(land, land_src) @ mom   (all npt rows)
  {
    int tilesPerBatch = npt / 16;            // 256
    int blocksPerBatch = tilesPerBatch / 8;  // 32
    rbf_step_kernel<<<B * blocksPerBatch, 256, 0, stream>>>(
        land, npt, /*iOffset=*/0,
        land, npt,
        momentum, n_src,
        shape1, npt,
        n_src, sigmaV2, blocksPerBatch);
  }

  // 3) step 2: out = shape1_tgt + TAU * K(shape1_tgt, shape1_src) @ mom
  {
    int tilesPerBatch = n_tgt / 16;          // 128
    int blocksPerBatch = tilesPerBatch / 8;  // 16
    rbf_step_kernel<<<B * blocksPerBatch, 256, 0, stream>>>(
        shape1, npt, /*iOffset=*/n_src,
        shape1, npt,
        momentum, n_src,
        out, n_tgt,
        n_src, sigmaV2, blocksPerBatch);
  }
}